// PVAE_84524956385425
// MI455X (gfx1250) — compile-verified
//
#include <hip/hip_runtime.h>

typedef __attribute__((ext_vector_type(2))) float v2f;
typedef __attribute__((ext_vector_type(4))) float v4f;
typedef __attribute__((ext_vector_type(8))) float v8f;

#define BSZ      4096
#define DIN      784
#define DZ       128
#define NTRIALS  256

// ---------------------------------------------------------------------------
// Encoder: du = min(x @ W_enc + b_enc, 5);  inv_rate = 1/(exp(min(du+pr,5))+1e-6)
// One wave per 16x16 output tile, V_WMMA_F32_16X16X4_F32 over K=784.
// 256 row-tiles x 8 col-tiles = 2048 full waves (EXEC all ones).
// ---------------------------------------------------------------------------
__global__ __launch_bounds__(256) void pvae_enc_wmma(
    const float* __restrict__ x, const float* __restrict__ W,
    const float* __restrict__ bias, const float* __restrict__ prior,
    float* __restrict__ du_out, float* __restrict__ inv_rate)
{
    const int wave    = (blockIdx.x * 256 + threadIdx.x) >> 5;
    const int lane    = threadIdx.x & 31;
    const int rowTile = wave >> 3;            // 0..255
    const int colTile = wave & 7;             // 0..7
    const int row0    = rowTile * 16;
    const int col0    = colTile * 16;

    // A frag (16x4 f32): lanes 0-15 -> M=lane, K={0,1}; lanes 16-31 -> K={2,3}
    const int m  = lane & 15;
    const int kk = (lane >> 4) * 2;
    const float* __restrict__ xrow = x + (size_t)(row0 + m) * DIN + kk;
    // B frag (4x16 f32): lanes 0-15 -> K={0,1}, N=lane&15; lanes 16-31 -> K={2,3}
    const float* __restrict__ wcol = W + col0 + m;

    v8f c = {};
    #pragma unroll 4
    for (int k = 0; k < DIN; k += 4) {
        v2f a, b;
        a.x = xrow[k];
        a.y = xrow[k + 1];
        b.x = wcol[(size_t)(k + kk)     * DZ];
        b.y = wcol[(size_t)(k + kk + 1) * DZ];
        c = __builtin_amdgcn_wmma_f32_16x16x4_f32(
                false, a, false, b, (short)0, c, false, false);
    }

    // C/D layout: VGPR r: lanes 0-15 -> M=r, lanes 16-31 -> M=r+8; N = lane&15
    const int col   = col0 + (lane & 15);
    const int rbase = row0 + ((lane >> 4) << 3);
    const float pr  = fminf(prior[col], 5.0f);
    const float be  = bias[col];
    #pragma unroll
    for (int r = 0; r < 8; ++r) {
        const int   row = rbase + r;
        const float du  = fminf(c[r] + be, 5.0f);
        du_out[(size_t)row * DZ + col] = du;
        const float lr   = fminf(du + pr, 5.0f);
        const float rate = __expf(lr) + 1e-6f;
        inv_rate[(size_t)row * DZ + col] = 1.0f / rate;
    }
}

// ---------------------------------------------------------------------------
// Exponential race (memory-bound stage: 512MB of u streamed once).
// Each thread owns 4 consecutive z channels: one nontemporal GLOBAL_LOAD_B128
// per trial slice (512B contiguous per wave), 4 independent (times,acc)
// chains for transcendental ILP, wave-uniform early exit once every chain in
// the wave has saturated (remaining sigmoid terms < 1.5e-8 each).
// ---------------------------------------------------------------------------
__global__ __launch_bounds__(256) void pvae_race(
    const float* __restrict__ u, const float* __restrict__ inv_rate,
    float* __restrict__ z_out)
{
    const int tid  = blockIdx.x * 256 + threadIdx.x;     // 0 .. 131071
    const int base = tid * 4;                            // element index, 16B aligned
    const size_t stride4 = (size_t)(BSZ * DZ) / 4;       // v4f elements per slice

    const v4f ir = *(const v4f*)(inv_rate + base);
    const v4f* __restrict__ up = (const v4f*)u + tid;

    float t0 = 0.f, t1 = 0.f, t2 = 0.f, t3 = 0.f;
    float a0 = 0.f, a1 = 0.f, a2 = 0.f, a3 = 0.f;

    for (int t = 0; t < NTRIALS; t += 2) {
        const v4f uA = __builtin_nontemporal_load(up + (size_t)t * stride4);
        const v4f uB = __builtin_nontemporal_load(up + (size_t)(t + 1) * stride4);

        t0 += -__logf(1.0f - uA.x) * ir.x;
        t1 += -__logf(1.0f - uA.y) * ir.y;
        t2 += -__logf(1.0f - uA.z) * ir.z;
        t3 += -__logf(1.0f - uA.w) * ir.w;
        a0 += 1.0f / (1.0f + __expf(t0 - 1.0f));
        a1 += 1.0f / (1.0f + __expf(t1 - 1.0f));
        a2 += 1.0f / (1.0f + __expf(t2 - 1.0f));
        a3 += 1.0f / (1.0f + __expf(t3 - 1.0f));

        t0 += -__logf(1.0f - uB.x) * ir.x;
        t1 += -__logf(1.0f - uB.y) * ir.y;
        t2 += -__logf(1.0f - uB.z) * ir.z;
        t3 += -__logf(1.0f - uB.w) * ir.w;
        a0 += 1.0f / (1.0f + __expf(t0 - 1.0f));
        a1 += 1.0f / (1.0f + __expf(t1 - 1.0f));
        a2 += 1.0f / (1.0f + __expf(t2 - 1.0f));
        a3 += 1.0f / (1.0f + __expf(t3 - 1.0f));

        // times monotone increasing; once every chain in the wave exceeds the
        // threshold, all remaining terms are < sigmoid(-18) ~ 1.5e-8 -> skip.
        const float tmin = fminf(fminf(t0, t1), fminf(t2, t3));
        if (__all(tmin > 19.0f)) break;
    }

    v4f zv; zv.x = a0; zv.y = a1; zv.z = a2; zv.w = a3;
    *((v4f*)(z_out) + tid) = zv;
}

// ---------------------------------------------------------------------------
// Decoder: y = sigmoid(z @ W_dec + b_dec). K=128 -> 32 WMMAs per tile.
// 256 row-tiles x 49 col-tiles = 12544 full waves (exact grid, no guards).
// ---------------------------------------------------------------------------
__global__ __launch_bounds__(256) void pvae_dec_wmma(
    const float* __restrict__ z, const float* __restrict__ W,
    const float* __restrict__ bias, float* __restrict__ y)
{
    const int wave    = (blockIdx.x * 256 + threadIdx.x) >> 5;
    const int lane    = threadIdx.x & 31;
    const int rowTile = wave / 49;
    const int colTile = wave % 49;
    const int row0    = rowTile * 16;
    const int col0    = colTile * 16;

    const int m  = lane & 15;
    const int kk = (lane >> 4) * 2;
    const float* __restrict__ zrow = z + (size_t)(row0 + m) * DZ + kk;
    const float* __restrict__ wcol = W + col0 + m;

    v8f c = {};
    #pragma unroll
    for (int k = 0; k < DZ; k += 4) {
        v2f a, b;
        a.x = zrow[k];
        a.y = zrow[k + 1];
        b.x = wcol[(size_t)(k + kk)     * DIN];
        b.y = wcol[(size_t)(k + kk + 1) * DIN];
        c = __builtin_amdgcn_wmma_f32_16x16x4_f32(
                false, a, false, b, (short)0, c, false, false);
    }

    const int col   = col0 + (lane & 15);
    const int rbase = row0 + ((lane >> 4) << 3);
    const float be  = bias[col];
    #pragma unroll
    for (int r = 0; r < 8; ++r) {
        const int   row = rbase + r;
        const float v   = c[r] + be;
        y[(size_t)row * DIN + col] = 1.0f / (1.0f + __expf(-v));
    }
}

// ---------------------------------------------------------------------------
// d_in order: x, u, W_enc, b_enc, W_dec, b_dec, prior
// d_out: du[4096*128] | z[4096*128] | y[4096*784]
// d_ws:  inv_rate[4096*128] f32 (2 MB)
// ---------------------------------------------------------------------------
extern "C" void kernel_launch(void* const* d_in, const int* in_sizes, int n_in,
                              void* d_out, int out_size, void* d_ws, size_t ws_size,
                              hipStream_t stream)
{
    const float* x      = (const float*)d_in[0];
    const float* u      = (const float*)d_in[1];
    const float* W_enc  = (const float*)d_in[2];
    const float* b_enc  = (const float*)d_in[3];
    const float* W_dec  = (const float*)d_in[4];
    const float* b_dec  = (const float*)d_in[5];
    const float* prior  = (const float*)d_in[6];

    float* out      = (float*)d_out;
    float* du       = out;                         // [4096,128]
    float* zbuf     = out + (size_t)BSZ * DZ;      // [4096,128]
    float* y        = out + (size_t)2 * BSZ * DZ;  // [4096,784]
    float* inv_rate = (float*)d_ws;                // [4096,128]

    // Encoder: 2048 waves -> 256 blocks of 8 waves
    pvae_enc_wmma<<<256, 256, 0, stream>>>(x, W_enc, b_enc, prior, du, inv_rate);
    // Race: 131072 threads (4 z-channels each) -> 512 blocks
    pvae_race<<<512, 256, 0, stream>>>(u, inv_rate, zbuf);
    // Decoder: 12544 waves -> 1568 blocks of 8 waves
    pvae_dec_wmma<<<1568, 256, 0, stream>>>(zbuf, W_dec, b_dec, y);
}